// NodeGNBlock_88837103551520
// MI455X (gfx1250) — compile-verified
//
#include <hip/hip_runtime.h>

typedef float v2f __attribute__((ext_vector_type(2)));
typedef float v8f __attribute__((ext_vector_type(8)));

#define N_NODES 50000
#define N_EDGES 800000
#define D_NODE  64
#define D_EDGE  32
#define D_OUT   64
#define D_IN_E  160   // 64 + 32 + 64
#define D_IN_V  128   // 64 + 64

// LDS row strides padded to even (for ds_load_b64 alignment) and chosen so that
// stride mod 64 banks avoids systematic conflicts across the 16 lanes.
#define SXE_STRIDE 162
#define SWE_STRIDE 162
#define SXV_STRIDE 130
#define SWV_STRIDE 130

#define TILES_PER_BLOCK 8

__device__ __forceinline__ v8f wmma_f32_k4(v2f a, v2f b, v8f c) {
  // D = A(16x4, f32) * B(4x16, f32) + C(16x16, f32)
  return __builtin_amdgcn_wmma_f32_16x16x4_f32(
      /*neg_a=*/false, a, /*neg_b=*/false, b,
      /*c_mod=*/(short)0, c, /*reuse_a=*/false, /*reuse_b=*/false);
}

__global__ void zero_kernel(float* __restrict__ p, int n) {
  int i = blockIdx.x * blockDim.x + threadIdx.x;
  if (i < n) p[i] = 0.0f;
}

// ---------------------------------------------------------------------------
// Edge MLP: m = relu([nf[src] | ef | nf[dst]] @ W_e + b_e); atomic mean-prep:
//   msum[dst] += m ; deg[dst] += 1
// Block = 128 threads (4 waves). Each wave computes a 16x16 tile:
//   rows  = 16 edges, cols = one 16-wide slice of the 64 output channels.
// ---------------------------------------------------------------------------
__global__ __launch_bounds__(128) void edge_mlp_kernel(
    const float* __restrict__ nf, const float* __restrict__ ef,
    const int* __restrict__ src, const int* __restrict__ dst,
    const float* __restrict__ We, const float* __restrict__ be,
    float* __restrict__ msum, float* __restrict__ deg) {
  __shared__ float swt[D_OUT * SWE_STRIDE];   // W_e transposed: [n][k]
  __shared__ float sx[16 * SXE_STRIDE];       // gathered x_e tile: [row][k]
  __shared__ int   sidx[32];                  // [0..15]=src idx, [16..31]=dst idx
  __shared__ float sb[D_OUT];

  const int tid = threadIdx.x;

  // Stage transposed weights + bias (reused for all tiles of this block).
  for (int i = tid; i < D_IN_E * D_OUT; i += 128) {
    int k = i >> 6;       // row of W_e (input dim)
    int n = i & 63;       // col of W_e (output channel)
    swt[n * SWE_STRIDE + k] = We[i];
  }
  if (tid < D_OUT) sb[tid] = be[tid];

  const int lane = tid & 31;
  const int w    = tid >> 5;        // wave id -> N-tile
  const int hi   = lane >> 4;       // half-wave selects K pair / row+8
  const int ls   = lane & 15;
  const int col  = w * 16 + ls;     // output channel for B/D fragments

  for (int t = 0; t < TILES_PER_BLOCK; ++t) {
    const int tile = blockIdx.x * TILES_PER_BLOCK + t;
    const int e0 = tile * 16;

    __syncthreads();  // previous tile's compute done before restaging
    if (tid < 16)       sidx[tid] = src[e0 + tid];
    else if (tid < 32)  sidx[tid] = dst[e0 + (tid - 16)];
    __syncthreads();

    // Gather x_e tile into LDS: [16 edges][160 features]
    for (int i = tid; i < 16 * D_IN_E; i += 128) {
      int row = i / D_IN_E;
      int c   = i - row * D_IN_E;
      float v;
      if (c < D_NODE) {
        v = nf[(long)sidx[row] * D_NODE + c];
      } else if (c < D_NODE + D_EDGE) {
        v = ef[(long)(e0 + row) * D_EDGE + (c - D_NODE)];
      } else {
        v = nf[(long)sidx[16 + row] * D_NODE + (c - D_NODE - D_EDGE)];
      }
      sx[row * SXE_STRIDE + c] = v;
    }
    __syncthreads();

    v8f acc = {};
#pragma unroll
    for (int k0 = 0; k0 < D_IN_E; k0 += 4) {
      const int kk = k0 + 2 * hi;
      v2f a = *(const v2f*)&sx[ls * SXE_STRIDE + kk];     // A: row=ls, K pair
      v2f b = *(const v2f*)&swt[col * SWE_STRIDE + kk];   // B: col,   K pair
      acc = wmma_f32_k4(a, b, acc);
    }

    const float bias = sb[col];
#pragma unroll
    for (int r = 0; r < 8; ++r) {
      const int rowM = r + 8 * hi;                 // D layout: VGPR r, half-wave
      float v = acc[r] + bias;
      v = v > 0.0f ? v : 0.0f;                     // ReLU
      atomicAdd(&msum[(long)sidx[16 + rowM] * D_OUT + col], v);
    }
    if (w == 0 && lane < 16) {
      atomicAdd(&deg[sidx[16 + lane]], 1.0f);
    }
  }
}

// ---------------------------------------------------------------------------
// Node MLP: out = relu([msum/deg (0 if deg==0) | nf] @ W_v + b_v)
// ---------------------------------------------------------------------------
__global__ __launch_bounds__(128) void node_mlp_kernel(
    const float* __restrict__ nf, const float* __restrict__ msum,
    const float* __restrict__ deg, const float* __restrict__ Wv,
    const float* __restrict__ bv, float* __restrict__ out) {
  __shared__ float swt[D_OUT * SWV_STRIDE];   // W_v transposed: [n][k]
  __shared__ float sx[16 * SXV_STRIDE];       // x_v tile: [row][k]
  __shared__ float sb[D_OUT];

  const int tid = threadIdx.x;
  for (int i = tid; i < D_IN_V * D_OUT; i += 128) {
    int k = i >> 6;
    int n = i & 63;
    swt[n * SWV_STRIDE + k] = Wv[i];
  }
  if (tid < D_OUT) sb[tid] = bv[tid];

  const int lane = tid & 31;
  const int w    = tid >> 5;
  const int hi   = lane >> 4;
  const int ls   = lane & 15;
  const int col  = w * 16 + ls;
  const int n_tiles = N_NODES / 16;   // 3125

  for (int t = 0; t < TILES_PER_BLOCK; ++t) {
    const int tile = blockIdx.x * TILES_PER_BLOCK + t;
    if (tile >= n_tiles) break;       // uniform across block
    const int n0 = tile * 16;

    __syncthreads();
    for (int i = tid; i < 16 * D_IN_V; i += 128) {
      int row  = i >> 7;
      int c    = i & 127;
      int node = n0 + row;
      float v;
      if (c < D_OUT) {
        float d = deg[node];
        v = (d > 0.0f) ? msum[(long)node * D_OUT + c] / d : 0.0f;
      } else {
        v = nf[(long)node * D_NODE + (c - D_OUT)];
      }
      sx[row * SXV_STRIDE + c] = v;
    }
    __syncthreads();

    v8f acc = {};
#pragma unroll
    for (int k0 = 0; k0 < D_IN_V; k0 += 4) {
      const int kk = k0 + 2 * hi;
      v2f a = *(const v2f*)&sx[ls * SXV_STRIDE + kk];
      v2f b = *(const v2f*)&swt[col * SWV_STRIDE + kk];
      acc = wmma_f32_k4(a, b, acc);
    }

    const float bias = sb[col];
#pragma unroll
    for (int r = 0; r < 8; ++r) {
      const int rowM = r + 8 * hi;
      float v = acc[r] + bias;
      out[(long)(n0 + rowM) * D_OUT + col] = v > 0.0f ? v : 0.0f;
    }
  }
}

extern "C" void kernel_launch(void* const* d_in, const int* in_sizes, int n_in,
                              void* d_out, int out_size, void* d_ws, size_t ws_size,
                              hipStream_t stream) {
  const float* nf  = (const float*)d_in[0];
  const float* ef  = (const float*)d_in[1];
  const int*   src = (const int*)d_in[2];
  const int*   dst = (const int*)d_in[3];
  const float* We  = (const float*)d_in[4];
  const float* be  = (const float*)d_in[5];
  const float* Wv  = (const float*)d_in[6];
  const float* bv  = (const float*)d_in[7];

  float* msum = (float*)d_ws;                       // [N_NODES * D_OUT]
  float* deg  = msum + (size_t)N_NODES * D_OUT;     // [N_NODES]
  float* out  = (float*)d_out;

  // 1) zero accumulators (msum and deg are contiguous in d_ws)
  const int zn = N_NODES * D_OUT + N_NODES;
  zero_kernel<<<(zn + 255) / 256, 256, 0, stream>>>(msum, zn);

  // 2) edge MLP + atomic aggregation: 50000 tiles of 16 edges, 8 per block
  edge_mlp_kernel<<<N_EDGES / 16 / TILES_PER_BLOCK, 128, 0, stream>>>(
      nf, ef, src, dst, We, be, msum, deg);

  // 3) node MLP: 3125 tiles of 16 nodes, 8 per block
  const int node_blocks = (N_NODES / 16 + TILES_PER_BLOCK - 1) / TILES_PER_BLOCK;
  node_mlp_kernel<<<node_blocks, 128, 0, stream>>>(nf, msum, deg, Wv, bv, out);
}